// Segmenter_53051436040558
// MI455X (gfx1250) — compile-verified
//
#include <hip/hip_runtime.h>
#include <hip/hip_bf16.h>

// ---------------- problem constants ----------------
constexpr int kB   = 8;
constexpr int kTok = 576;
constexpr int kD   = 768;              // D_MODEL
constexpr int kK   = 256;              // K_PSI
constexpr int kC   = 150;              // N_CLS
constexpr int kM   = kB * kTok;        // 4608
constexpr int HF = 24, WF = 24, HI = 384, WI = 384;
constexpr int LDLOG = 160;             // padded logits leading dim (150 -> 160)
constexpr float FM_MOM = 0.99f;
constexpr float KM_MOM = 0.999f;
constexpr float T_NEF  = 10.0f;
constexpr float ALPHA  = 1.0f;
constexpr float EPSF   = 1e-12f;

typedef float v2f __attribute__((ext_vector_type(2)));
typedef float v8f __attribute__((ext_vector_type(8)));

// ---------------- helpers ----------------
__device__ __forceinline__ float block_reduce_sum(float v, float* sm) {
  int t = threadIdx.x;
  sm[t] = v; __syncthreads();
  for (int s = blockDim.x >> 1; s > 0; s >>= 1) {
    if (t < s) sm[t] += sm[t + s];
    __syncthreads();
  }
  float r = sm[0];
  __syncthreads();
  return r;
}

// One 16x16 f32 output tile, accumulated over kdim with V_WMMA_F32_16X16X4_F32.
// A element (row,k) at A[row*rsA + k*csA]; B element (k,col) at B[k*rsB + col*csB].
// colMax clamps B column loads (for N_CLS=150 padded to 160).
__device__ __forceinline__ v8f wmma_tile_f32(const float* __restrict__ A, long rsA, long csA,
                                             const float* __restrict__ Bm, long rsB, long csB,
                                             int kdim, int row0, int col0, int colMax) {
  const int lane = threadIdx.x & 31;
  const int m  = lane & 15;             // A row / B col within tile
  const int kh = (lane >> 4) << 1;      // 0 or 2 (K sub-offset per lane half)
  int col = col0 + m; if (col >= colMax) col = colMax - 1;
  const long rowBase = (long)(row0 + m) * rsA;
  const long colBase = (long)col * csB;
  v8f c = {};
  for (int k = 0; k < kdim; k += 4) {
    v2f a, b;
    a[0] = A[rowBase + (long)(k + kh)     * csA];
    a[1] = A[rowBase + (long)(k + kh + 1) * csA];
    b[0] = Bm[(long)(k + kh)     * rsB + colBase];
    b[1] = Bm[(long)(k + kh + 1) * rsB + colBase];
    c = __builtin_amdgcn_wmma_f32_16x16x4_f32(false, a, false, b, (short)0, c, false, false);
  }
  return c;
}

__device__ __forceinline__ void wmma_store(float* __restrict__ D, long ldD,
                                           int row0, int col0, int colMax,
                                           v8f c, float scale) {
  const int lane = threadIdx.x & 31;
  const int m  = lane & 15;
  const int rh = (lane >> 4) << 3;      // 0 or 8
  if (col0 + m < colMax) {
    for (int r = 0; r < 8; ++r)
      D[(long)(row0 + rh + r) * ldD + col0 + m] = c[r] * scale;
  }
}

// ---------------- small kernels ----------------
__global__ void zero_kernel(float* p, int n) {
  int i = blockIdx.x * blockDim.x + threadIdx.x;
  if (i < n) p[i] = 0.0f;
}

// fm[d] = FM_MOM*feature_mean[d] + (1-FM_MOM)*mean_m x[m,d]
__global__ void fm_kernel(const float* __restrict__ x, const float* __restrict__ fmean_in,
                          float* __restrict__ fm) {
  __shared__ float sm[256];
  int d = blockIdx.x;
  float acc = 0.f;
  for (int m = threadIdx.x; m < kM; m += 256) acc += x[(long)m * kD + d];
  float tot = block_reduce_sum(acc, sm);
  if (threadIdx.x == 0)
    fm[d] = FM_MOM * fmean_in[d] + (1.0f - FM_MOM) * (tot / (float)kM);
}

// fmn = fm / max(||fm||, eps)
__global__ void fmn_kernel(const float* __restrict__ fm, float* __restrict__ fmn) {
  __shared__ float sm[256];
  float acc = 0.f;
  for (int d = threadIdx.x; d < kD; d += 256) { float v = fm[d]; acc += v * v; }
  float ss = block_reduce_sum(acc, sm);
  float inv = 1.0f / fmaxf(sqrtf(ss), EPSF);
  for (int d = threadIdx.x; d < kD; d += 256) fmn[d] = fm[d] * inv;
}

// Psi = FF @ W_psi + b_psi  : [4608x768]x[768x256]
__global__ void psi_gemm_kernel(const float* __restrict__ FF, const float* __restrict__ W,
                                const float* __restrict__ bias, float* __restrict__ Psi) {
  int wid = (blockIdx.x * blockDim.x + threadIdx.x) >> 5;  // 4608 tiles: 288 x 16
  int tM = wid >> 4, tN = wid & 15;
  v8f c = wmma_tile_f32(FF, kD, 1, W, kK, 1, kD, tM * 16, tN * 16, kK);
  const int lane = threadIdx.x & 31;
  const int m = lane & 15, rh = (lane >> 4) << 3;
  float bv = bias[tN * 16 + m];
  for (int r = 0; r < 8; ++r)
    Psi[(long)(tM * 16 + rh + r) * kK + tN * 16 + m] = c[r] + bv;
}

// row-l2-normalize Psi -> Psin (one block of 256 per row)
__global__ void psin_kernel(const float* __restrict__ Psi, float* __restrict__ Psin) {
  __shared__ float sm[256];
  long row = blockIdx.x;
  float v = Psi[row * kK + threadIdx.x];
  float ss = block_reduce_sum(v * v, sm);
  float inv = 1.0f / fmaxf(sqrtf(ss), EPSF);
  Psin[row * kK + threadIdx.x] = v * inv;
}

// csq[n] = ||C_n||^2
__global__ void csq_kernel(const float* __restrict__ C, float* __restrict__ csq) {
  __shared__ float sm[256];
  int n = blockIdx.x;
  float v = C[(long)n * kK + threadIdx.x];
  float ss = block_reduce_sum(v * v, sm);
  if (threadIdx.x == 0) csq[n] = ss;
}

// logits[m,n] = -(1 + csq[n] - 2 * Psin_m . C_n) ; padded cols masked on store
__global__ void logits_gemm_kernel(const float* __restrict__ Psin, const float* __restrict__ C,
                                   const float* __restrict__ csq, float* __restrict__ logits) {
  int wid = (blockIdx.x * blockDim.x + threadIdx.x) >> 5;  // 2880 tiles: 288 x 10
  int tM = wid / 10, tN = wid % 10;
  v8f acc = wmma_tile_f32(Psin, kK, 1, C, 1, kK, kK, tM * 16, tN * 16, kC);
  const int lane = threadIdx.x & 31;
  const int m = lane & 15, rh = (lane >> 4) << 3;
  int col = tN * 16 + m;
  if (col < kC) {
    float cs = csq[col];
    for (int r = 0; r < 8; ++r)
      logits[(long)(tM * 16 + rh + r) * LDLOG + col] = -(1.0f + cs - 2.0f * acc[r]);
  }
}

// per-row argmax (first-index tiebreak) + scatter-add of Psin row into centers acc
__global__ void argmax_kernel(const float* __restrict__ logits, const float* __restrict__ Psin,
                              float* __restrict__ counts, float* __restrict__ cacc) {
  int row  = (blockIdx.x * blockDim.x + threadIdx.x) >> 5;  // one wave per row
  int lane = threadIdx.x & 31;
  float best = -__builtin_inff(); int bidx = 0x7fffffff;
  for (int n = lane; n < kC; n += 32) {
    float v = logits[(long)row * LDLOG + n];
    if (v > best || (v == best && n < bidx)) { best = v; bidx = n; }
  }
  for (int off = 16; off > 0; off >>= 1) {
    float ov = __shfl_xor(best, off, 32);
    int   oi = __shfl_xor(bidx, off, 32);
    if (ov > best || (ov == best && oi < bidx)) { best = ov; bidx = oi; }
  }
  if (lane == 0) atomicAdd(&counts[bidx], 1.0f);
  for (int k = lane; k < kK; k += 32)
    atomicAdd(&cacc[(long)bidx * kK + k], Psin[(long)row * kK + k]);
}

// new_centers = KM_MOM*C + (1-KM_MOM)*cacc ; new_num = num + counts
__global__ void centers_kernel(const float* __restrict__ C, const float* __restrict__ cacc,
                               const float* __restrict__ counts, const float* __restrict__ num_in,
                               float* __restrict__ out_centers, float* __restrict__ out_num) {
  int n = blockIdx.x, t = threadIdx.x;
  long i = (long)n * kK + t;
  out_centers[i] = KM_MOM * C[i] + (1.0f - KM_MOM) * cacc[i];
  if (t == 0) out_num[n] = num_in[n] + counts[n];
}

// bilinear 24x24 -> 384x384 upsample (half-pixel centers, edge clamp)
__global__ void masks_kernel(const float* __restrict__ logits, float* __restrict__ out) {
  long idx = (long)blockIdx.x * blockDim.x + threadIdx.x;
  int x = (int)(idx % WI);
  long t = idx / WI;
  int y = (int)(t % HI); t /= HI;
  int c = (int)(t % kC);
  int b = (int)(t / kC);
  float fy = (y + 0.5f) * (1.0f / 16.0f) - 0.5f;
  float fx = (x + 0.5f) * (1.0f / 16.0f) - 0.5f;
  int y0 = (int)floorf(fy), x0 = (int)floorf(fx);
  float wy = fy - (float)y0, wx = fx - (float)x0;
  int y0c = min(max(y0, 0), HF - 1), y1c = min(max(y0 + 1, 0), HF - 1);
  int x0c = min(max(x0, 0), WF - 1), x1c = min(max(x0 + 1, 0), WF - 1);
  const float* Lb = logits + (long)b * kTok * LDLOG + c;
  float v00 = Lb[(long)(y0c * WF + x0c) * LDLOG];
  float v01 = Lb[(long)(y0c * WF + x1c) * LDLOG];
  float v10 = Lb[(long)(y1c * WF + x0c) * LDLOG];
  float v11 = Lb[(long)(y1c * WF + x1c) * LDLOG];
  float v0 = v00 + (v01 - v00) * wx;
  float v1 = v10 + (v11 - v10) * wx;
  out[idx] = v0 + (v1 - v0) * wy;
}

// Q = diag(d) * FFn, d = mask_neg_rsqrt(FFn . fmn)  (one block per row)
__global__ void q_kernel(const float* __restrict__ FF, const float* __restrict__ fmn,
                         float* __restrict__ Q) {
  __shared__ float sm[256];
  long row = blockIdx.x;
  int t = threadIdx.x;
  float v0 = FF[row * kD + t], v1 = FF[row * kD + t + 256], v2 = FF[row * kD + t + 512];
  float f0 = fmn[t], f1 = fmn[t + 256], f2 = fmn[t + 512];
  float ss = block_reduce_sum(v0 * v0 + v1 * v1 + v2 * v2, sm);
  float dt = block_reduce_sum(v0 * f0 + v1 * f1 + v2 * f2, sm);
  float inv  = 1.0f / fmaxf(sqrtf(ss), EPSF);
  float dotn = dt * inv;
  float dfac = (dotn > 0.0f) ? rsqrtf(dotn) : 0.0f;
  float sc = inv * dfac;
  Q[row * kD + t] = v0 * sc;
  Q[row * kD + t + 256] = v1 * sc;
  Q[row * kD + t + 512] = v2 * sc;
}

// T1 = s * (Q^T @ Psi) : [768x4608]x[4608x256], s = T_NEF/M (one PsiL factor)
__global__ void t1_gemm_kernel(const float* __restrict__ Q, const float* __restrict__ Psi,
                               float* __restrict__ T1) {
  int wid = (blockIdx.x * blockDim.x + threadIdx.x) >> 5;  // 768 tiles: 48 x 16
  int tM = wid >> 4, tN = wid & 15;
  v8f c = wmma_tile_f32(Q, 1, kD, Psi, kK, 1, kM, tM * 16, tN * 16, kK);
  wmma_store(T1, kK, tM * 16, tN * 16, kK, c, T_NEF / (float)kM);
}

// GP = Q @ T1 : [4608x768]x[768x256]
__global__ void gp_gemm_kernel(const float* __restrict__ Q, const float* __restrict__ T1,
                               float* __restrict__ GP) {
  int wid = (blockIdx.x * blockDim.x + threadIdx.x) >> 5;  // 4608 tiles: 288 x 16
  int tM = wid >> 4, tN = wid & 15;
  v8f c = wmma_tile_f32(Q, kD, 1, T1, kK, 1, kD, tM * 16, tN * 16, kK);
  wmma_store(GP, kK, tM * 16, tN * 16, kK, c, 1.0f);
}

// R = s * (Psi^T @ GP) : [256x4608]x[4608x256]  (second PsiL factor)
__global__ void r_gemm_kernel(const float* __restrict__ Psi, const float* __restrict__ GP,
                              float* __restrict__ R) {
  int wid = (blockIdx.x * blockDim.x + threadIdx.x) >> 5;  // 256 tiles: 16 x 16
  int tM = wid >> 4, tN = wid & 15;
  v8f c = wmma_tile_f32(Psi, 1, kK, GP, kK, 1, kM, tM * 16, tN * 16, kK);
  wmma_store(R, kK, tM * 16, tN * 16, kK, c, T_NEF / (float)kM);
}

// loss = -2*trace(R); reg = ALPHA * sum triu(R^2, k=1)
__global__ void loss_reg_kernel(const float* __restrict__ R, float* __restrict__ out_loss,
                                float* __restrict__ out_reg) {
  __shared__ float sm[256];
  float tr = 0.f, tu = 0.f;
  for (int i = threadIdx.x; i < kK * kK; i += 256) {
    int r = i / kK, c = i % kK;
    float v = R[i];
    if (r == c) tr += v;
    else if (c > r) tu += v * v;
  }
  float trs = block_reduce_sum(tr, sm);
  float tus = block_reduce_sum(tu, sm);
  if (threadIdx.x == 0) { *out_loss = -2.0f * trs; *out_reg = ALPHA * tus; }
}

// ---------------- launcher ----------------
extern "C" void kernel_launch(void* const* d_in, const int* in_sizes, int n_in,
                              void* d_out, int out_size, void* d_ws, size_t ws_size,
                              hipStream_t stream) {
  const float* x     = (const float*)d_in[0];   // [8,576,768] == FF [4608,768]
  const float* W_psi = (const float*)d_in[1];   // [768,256]
  const float* b_psi = (const float*)d_in[2];   // [256]
  const float* Cctr  = (const float*)d_in[3];   // [150,256]
  const float* num_in= (const float*)d_in[4];   // [150]
  const float* fmean = (const float*)d_in[5];   // [768]
  float* out = (float*)d_out;

  // output layout (flat, return order)
  constexpr long MASK_N = (long)kB * kC * HI * WI;        // 176,947,200
  float* out_masks   = out;
  float* out_loss    = out + MASK_N;
  float* out_reg     = out + MASK_N + 1;
  float* out_centers = out + MASK_N + 2;
  float* out_num     = out + MASK_N + 2 + (long)kC * kK;

  // workspace layout
  float* ws     = (float*)d_ws;
  float* fm     = ws;                       // 768
  float* fmn    = fm + 768;                 // 768
  float* csq    = fmn + 768;                // 160
  float* counts = csq + 160;                // 160   (zeroed, contiguous with cacc)
  float* cacc   = counts + 160;             // 150*256
  float* Psi    = cacc + (long)kC * kK;     // 4608*256
  float* Psin   = Psi  + (long)kM * kK;     // 4608*256
  float* logits = Psin + (long)kM * kK;     // 4608*160
  float* Q      = logits + (long)kM * LDLOG;// 4608*768
  float* T1     = Q    + (long)kM * kD;     // 768*256
  float* GP     = T1   + (long)kD * kK;     // 4608*256
  float* R      = GP   + (long)kM * kK;     // 256*256

  // 1) zero atomic accumulators (counts + cacc are contiguous)
  {
    int n = 160 + kC * kK;
    zero_kernel<<<(n + 255) / 256, 256, 0, stream>>>(counts, n);
  }
  // 2) feature-mean EMA + its normalization
  fm_kernel <<<kD, 256, 0, stream>>>(x, fmean, fm);
  fmn_kernel<<<1,  256, 0, stream>>>(fm, fmn);
  // 3) Psi head GEMM (WMMA f32)
  psi_gemm_kernel<<<(kM / 16) * (kK / 16) / 8, 256, 0, stream>>>(x, W_psi, b_psi, Psi);
  // 4) normalize Psi rows; class sq-norms
  psin_kernel<<<kM, 256, 0, stream>>>(Psi, Psin);
  csq_kernel <<<kC, 256, 0, stream>>>(Cctr, csq);
  // 5) logits = -(cdist^2) via WMMA dot + epilogue
  logits_gemm_kernel<<<(kM / 16) * 10 / 8, 256, 0, stream>>>(Psin, Cctr, csq, logits);
  // 6) argmax + scatter EMA stats, then finalize centers / counts
  argmax_kernel <<<kM / 8, 256, 0, stream>>>(logits, Psin, counts, cacc);
  centers_kernel<<<kC, 256, 0, stream>>>(Cctr, cacc, counts, num_in, out_centers, out_num);
  // 7) masks: bilinear 24->384 upsample (HBM-bound: ~708 MB out)
  masks_kernel<<<(unsigned)(MASK_N / 256), 256, 0, stream>>>(logits, out_masks);
  // 8) neuralef: G = Q Q^T factorization, GP = Q (Q^T PsiL)
  q_kernel      <<<kM, 256, 0, stream>>>(x, fmn, Q);
  t1_gemm_kernel<<<(kD / 16) * (kK / 16) / 8, 256, 0, stream>>>(Q, Psi, T1);
  gp_gemm_kernel<<<(kM / 16) * (kK / 16) / 8, 256, 0, stream>>>(Q, T1, GP);
  r_gemm_kernel <<<(kK / 16) * (kK / 16) / 8, 256, 0, stream>>>(Psi, GP, R);
  // 9) scalars
  loss_reg_kernel<<<1, 256, 0, stream>>>(R, out_loss, out_reg);
}